// EdgeSelectionRL_53085795779479
// MI455X (gfx1250) — compile-verified
//
#include <hip/hip_runtime.h>
#include <math.h>

typedef __attribute__((ext_vector_type(2))) float v2f;
typedef __attribute__((ext_vector_type(8))) float v8f;

#define B_ 8
#define C_ 256
#define F_ 128
#define H_ 256

// ---------------------------------------------------------------------------
// Kernel 1: a'[b,i,h] = xa[b,i,:]·W1[h,0:F] + b1[h]
//           c [b,j,h] = xa[b,j,:]·W1[h,F:2F]
// One wave per 16(i/j) x 16(h) tile, V_WMMA_F32_16X16X4_F32, K-loop over F.
// A frag (16x4 f32): lanes 0-15 hold row M=lane, K={k,k+1}; lanes 16-31 K={k+2,k+3}.
// B frag (4x16 f32): lanes 0-15 hold col N=lane, K={k,k+1}; lanes 16-31 K={k+2,k+3}.
// C/D (16x16 f32):  vgpr r: M = r + 8*(lane>>4), N = lane&15.
// ---------------------------------------------------------------------------
__global__ __launch_bounds__(128) void gemm_ac_wmma(
    const float* __restrict__ xa, const float* __restrict__ W1,
    const float* __restrict__ b1,
    float* __restrict__ ws_a, float* __restrict__ ws_c)
{
  const int wave = blockIdx.x * (blockDim.x >> 5) + (threadIdx.x >> 5);
  const int lane = threadIdx.x & 31;

  int t = wave;
  const int ht  = t & 15; t >>= 4;   // h tile  (H/16 = 16)
  const int it  = t & 15; t >>= 4;   // i tile  (C/16 = 16)
  const int b   = t & 7;  t >>= 3;   // batch
  const int sel = t & 1;             // 0 -> a (Wa), 1 -> c (Wb)

  const int i0 = it * 16;
  const int h0 = ht * 16;
  const int lo = lane & 15;          // M for A / N for B
  const int hi = lane >> 4;          // selects K pair

  const float* arow = xa + ((size_t)(b * C_ + i0 + lo)) * F_;
  const float* brow = W1 + (size_t)(h0 + lo) * (2 * F_) + sel * F_;

  v8f acc = {};
#pragma unroll 8
  for (int k = 0; k < F_; k += 4) {
    const int ka = k + 2 * hi;                 // even -> 8B aligned
    v2f af = *(const v2f*)(arow + ka);
    v2f bf = *(const v2f*)(brow + ka);
    // 8 args: (neg_a, A, neg_b, B, c_mod, C, reuse_a, reuse_b)
    acc = __builtin_amdgcn_wmma_f32_16x16x4_f32(false, af, false, bf,
                                                (short)0, acc, false, false);
  }

  const float bias = (sel == 0) ? b1[h0 + lo] : 0.0f;
  float* __restrict__ dst = (sel == 0) ? ws_a : ws_c;
#pragma unroll
  for (int r = 0; r < 8; ++r) {
    const int m = i0 + r + 8 * hi;
    dst[((size_t)(b * C_ + m)) * H_ + (h0 + lo)] = acc[r] + bias;
  }
}

// ---------------------------------------------------------------------------
// Kernel 2: out[b,i,j] = sigmoid( sum_h relu(a'[b,i,h] + c[b,j,h]) * w2[h] + b2 )
// 32x32 (i,j) tile per 256-thread block. a-tile: row-major LDS (wave-uniform
// broadcast reads). c-tile: transposed [h][j] with pad 33 -> lane-consecutive,
// bank-conflict-free reads. Each wave handles 4 i rows x 32 j lanes.
// ---------------------------------------------------------------------------
__global__ __launch_bounds__(256) void edge_epilogue(
    const float* __restrict__ ws_a, const float* __restrict__ ws_c,
    const float* __restrict__ w2, const float* __restrict__ b2p,
    float* __restrict__ out)
{
  __shared__ float a_s[32 * H_];      // [i][h]        32 KB
  __shared__ float c_s[H_ * 33];      // [h][j] padded 33 KB
  __shared__ float w_s[H_];           //                1 KB

  const int tid = threadIdx.x;
  int t = blockIdx.x;
  const int jt = t & 7; t >>= 3;
  const int it = t & 7; t >>= 3;
  const int b  = t;                   // 0..7
  const int i0 = it * 32, j0 = jt * 32;

  // a tile: 32 contiguous rows of H -> one contiguous 32 KB chunk
  {
    const float4* src = (const float4*)(ws_a + ((size_t)(b * C_ + i0)) * H_);
    float4* dst4 = (float4*)a_s;
#pragma unroll
    for (int idx = tid; idx < (32 * H_) / 4; idx += 256) dst4[idx] = src[idx];
  }
  // c tile: read coalesced along h, scatter transposed into LDS
  {
    const float4* src = (const float4*)(ws_c + ((size_t)(b * C_ + j0)) * H_);
#pragma unroll
    for (int idx = tid; idx < (32 * H_) / 4; idx += 256) {
      const int j  = idx >> 6;        // H/4 = 64 quads per j row
      const int hb = (idx & 63) * 4;
      float4 v = src[idx];
      c_s[(hb + 0) * 33 + j] = v.x;
      c_s[(hb + 1) * 33 + j] = v.y;
      c_s[(hb + 2) * 33 + j] = v.z;
      c_s[(hb + 3) * 33 + j] = v.w;
    }
  }
  if (tid < H_) w_s[tid] = w2[tid];
  __syncthreads();

  const int ti = tid >> 5;            // wave id 0..7 -> i group
  const int tj = tid & 31;            // lane -> j
  const float* a0 = a_s + (ti +  0) * H_;
  const float* a1 = a_s + (ti +  8) * H_;
  const float* a2 = a_s + (ti + 16) * H_;
  const float* a3 = a_s + (ti + 24) * H_;

  float acc0 = 0.f, acc1 = 0.f, acc2 = 0.f, acc3 = 0.f;
#pragma unroll 4
  for (int h = 0; h < H_; ++h) {
    const float wv = w_s[h];               // broadcast
    const float cv = c_s[h * 33 + tj];     // conflict-free
    float x;
    x = a0[h] + cv; acc0 = fmaf(fmaxf(x, 0.f), wv, acc0);
    x = a1[h] + cv; acc1 = fmaf(fmaxf(x, 0.f), wv, acc1);
    x = a2[h] + cv; acc2 = fmaf(fmaxf(x, 0.f), wv, acc2);
    x = a3[h] + cv; acc3 = fmaf(fmaxf(x, 0.f), wv, acc3);
  }

  const float b2 = b2p[0];
  const size_t obase = ((size_t)b * C_) * C_;
  const int j = j0 + tj;
  float* o = out + obase + (size_t)(i0 + ti) * C_ + j;
  o[ 0 * 8 * C_] = 1.0f / (1.0f + __expf(-(acc0 + b2)));
  o[ 1 * 8 * C_] = 1.0f / (1.0f + __expf(-(acc1 + b2)));
  o[ 2 * 8 * C_] = 1.0f / (1.0f + __expf(-(acc2 + b2)));
  o[ 3 * 8 * C_] = 1.0f / (1.0f + __expf(-(acc3 + b2)));
}

// ---------------------------------------------------------------------------
extern "C" void kernel_launch(void* const* d_in, const int* in_sizes, int n_in,
                              void* d_out, int out_size, void* d_ws, size_t ws_size,
                              hipStream_t stream) {
  const float* xa = (const float*)d_in[0];   // (B,C,F)
  const float* W1 = (const float*)d_in[1];   // (H,2F)
  const float* b1 = (const float*)d_in[2];   // (H,)
  const float* w2 = (const float*)d_in[3];   // (H,)
  const float* b2 = (const float*)d_in[4];   // scalar
  float* out = (float*)d_out;                // (B,C,C)

  float* ws_a = (float*)d_ws;                       // B*C*H floats (2 MB)
  float* ws_c = ws_a + (size_t)B_ * C_ * H_;        // B*C*H floats (2 MB)

  // Stage 1: 2 * B * (C/16) * (H/16) = 4096 waves, 4 waves/block
  gemm_ac_wmma<<<1024, 128, 0, stream>>>(xa, W1, b1, ws_a, ws_c);

  // Stage 2: B * (C/32)^2 = 512 blocks of 256 threads
  edge_epilogue<<<512, 256, 0, stream>>>(ws_a, ws_c, w2, b2, out);
}